// RQ_VAE_66760971649618
// MI455X (gfx1250) — compile-verified
//
#include <hip/hip_runtime.h>
#include <math.h>

typedef float v2f __attribute__((ext_vector_type(2)));
typedef float v8f __attribute__((ext_vector_type(8)));

#define BDIM 8192
#define INDIM 768
#define DDIM 128
#define KCB 8192
#define NLAYERS 3
#define CWF 0.25f
#define NSEG 8

// D = A(16x4) * B(4x16) + C, f32 WMMA. acc-chained over 32 k-slices for K=128.
__device__ __forceinline__ v8f wmma4(v2f a, v2f b, v8f c) {
  return __builtin_amdgcn_wmma_f32_16x16x4_f32(false, a, false, b, (short)0, c,
                                               false, false);
}

// ---------------------------------------------------------------------------
// C[M,N] = act(A[M,K] @ W[N,K]^T + bias)   (torch Linear convention)
// grid = (M/128, N/16), block = 256 (8 waves; wave handles 16 rows x 16 cols)
// act: 0 = none, 1 = silu, 2 = sigmoid
// ---------------------------------------------------------------------------
__global__ void mlp_gemm(const float* __restrict__ A, const float* __restrict__ W,
                         const float* __restrict__ bias, float* __restrict__ C,
                         int K, int N, int act) {
  __shared__ __align__(16) float Ws[16 * 128];
  const int tid = threadIdx.x;
  const int wave = tid >> 5, lane = tid & 31;
  const int m16 = lane & 15;            // row within 16-tile (A) / col (B)
  const int khalf = (lane >> 4) * 2;    // k sub-offset per WMMA f32 layout
  const int rowBase = blockIdx.x * 128 + wave * 16;
  const int n0 = blockIdx.y * 16;

  v8f acc = {};
  const int nkc = K >> 7;
  for (int kc = 0; kc < nkc; ++kc) {
    __syncthreads();
    {
      // stage W[n0..n0+15][kc*128..+128) -> LDS, 8 floats per thread
      const int f = tid * 8;
      const int r = f >> 7, c = f & 127;
      const float* src = W + (size_t)(n0 + r) * K + kc * 128 + c;
      *(float4*)(Ws + f)     = *(const float4*)(src);
      *(float4*)(Ws + f + 4) = *(const float4*)(src + 4);
    }
    __syncthreads();
    const float* Arow = A + (size_t)(rowBase + m16) * K + kc * 128 + khalf;
#pragma unroll
    for (int s = 0; s < 32; ++s) {
      v2f a = *(const v2f*)(Arow + s * 4);
      v2f b = *(const v2f*)(Ws + m16 * 128 + s * 4 + khalf);
      acc = wmma4(a, b, acc);
    }
  }

  const int n = n0 + m16;
  const float bv = bias ? bias[n] : 0.f;
#pragma unroll
  for (int r = 0; r < 8; ++r) {
    const int row = rowBase + r + (lane >> 4) * 8;   // C/D layout: M = r + half*8
    float v = acc[r] + bv;
    if (act == 1)      v = v / (1.f + expf(-v));     // silu
    else if (act == 2) v = 1.f / (1.f + expf(-v));   // sigmoid
    C[(size_t)row * N + n] = v;
  }
}

// cbsq[k] = sum_d cb[k][d]^2 ; wave per row, grid = K/8, block 256
__global__ void row_sq(const float* __restrict__ cb, float* __restrict__ cbsq) {
  const int wave = threadIdx.x >> 5, lane = threadIdx.x & 31;
  const int k = blockIdx.x * 8 + wave;
  const float4 v = *(const float4*)(cb + (size_t)k * DDIM + lane * 4);
  float s = v.x * v.x + v.y * v.y + v.z * v.z + v.w * v.w;
#pragma unroll
  for (int m = 16; m; m >>= 1) s += __shfl_xor(s, m, 32);
  if (lane == 0) cbsq[k] = s;
}

// ---------------------------------------------------------------------------
// Fused distance GEMM + running argmin:  score = ||cb_n||^2 - 2 * res_m . cb_n
// (||res||^2 is row-constant -> irrelevant for argmin). Partial results per
// N-segment. grid = (B/128, NSEG), block = 256.
// ---------------------------------------------------------------------------
__global__ void dist_argmin(const float* __restrict__ res, const float* __restrict__ cb,
                            const float* __restrict__ cbsq,
                            float* __restrict__ pscore, int* __restrict__ pidx) {
  __shared__ __align__(16) float Bs[16 * 128];
  __shared__ float Bq[16];
  const int tid = threadIdx.x;
  const int wave = tid >> 5, lane = tid & 31;
  const int m16 = lane & 15;
  const int khalf = (lane >> 4) * 2;
  const int rowBase = blockIdx.x * 128 + wave * 16;
  const int seg = blockIdx.y;

  // keep this wave's 16x128 A tile (res rows) resident in VGPRs
  v2f a[32];
  const float* Arow = res + (size_t)(rowBase + m16) * DDIM + khalf;
#pragma unroll
  for (int s = 0; s < 32; ++s) a[s] = *(const v2f*)(Arow + s * 4);

  float best[8];
  int bidx[8];
#pragma unroll
  for (int r = 0; r < 8; ++r) { best[r] = 3.0e38f; bidx[r] = 0; }

  const int ntiles = KCB / 16 / NSEG;   // 64
  const int t0 = seg * ntiles;
  for (int t = 0; t < ntiles; ++t) {
    const int n0 = (t0 + t) * 16;
    __syncthreads();
    {
      const int f = tid * 8;
      const int r = f >> 7, c = f & 127;
      const float* src = cb + (size_t)(n0 + r) * DDIM + c;
      *(float4*)(Bs + f)     = *(const float4*)(src);
      *(float4*)(Bs + f + 4) = *(const float4*)(src + 4);
      if (tid < 16) Bq[tid] = cbsq[n0 + tid];
    }
    __syncthreads();

    v8f acc = {};
#pragma unroll
    for (int s = 0; s < 32; ++s) {
      v2f b = *(const v2f*)(Bs + m16 * 128 + s * 4 + khalf);
      acc = wmma4(a[s], b, acc);
    }

    const float cq = Bq[m16];
    const int ncol = n0 + m16;
#pragma unroll
    for (int r = 0; r < 8; ++r) {
      const float sc = fmaf(-2.f, acc[r], cq);
      if (sc < best[r]) { best[r] = sc; bidx[r] = ncol; }   // strict '<' keeps first
    }
  }

  // reduce over 16 lanes within each half (same rows, different n columns)
#pragma unroll
  for (int r = 0; r < 8; ++r) {
#pragma unroll
    for (int m = 1; m <= 8; m <<= 1) {
      const float os = __shfl_xor(best[r], m, 32);
      const int oi = __shfl_xor(bidx[r], m, 32);
      if (os < best[r] || (os == best[r] && oi < bidx[r])) { best[r] = os; bidx[r] = oi; }
    }
    if (m16 == 0) {
      const int row = rowBase + r + (lane >> 4) * 8;
      pscore[(size_t)seg * BDIM + row] = best[r];
      pidx[(size_t)seg * BDIM + row] = bidx[r];
    }
  }
}

__global__ void argmin_fin(const float* __restrict__ pscore, const int* __restrict__ pidx,
                           int* __restrict__ sids, int l) {
  const int b = blockIdx.x * blockDim.x + threadIdx.x;
  float bs = 3.4e38f;
  int bi = 0;
  for (int s = 0; s < NSEG; ++s) {
    const float v = pscore[s * BDIM + b];
    const int i = pidx[s * BDIM + b];
    if (v < bs || (v == bs && i < bi)) { bs = v; bi = i; }
  }
  sids[b * NLAYERS + l] = bi;
}

// gather emb = cb[id]; qloss += 1.25*||res-emb||^2 ; embs_norm out ; res -= emb ; z += emb
// wave per row, grid = B/8, block 256
__global__ void vq_update(float* __restrict__ res, const float* __restrict__ cb,
                          const int* __restrict__ sids, float* __restrict__ z,
                          float* __restrict__ qloss, float* __restrict__ out, int l) {
  const int wave = threadIdx.x >> 5, lane = threadIdx.x & 31;
  const int b = blockIdx.x * 8 + wave;
  const int id = sids[b * NLAYERS + l];
  const size_t off = (size_t)b * DDIM + lane * 4;
  float4 rv = *(const float4*)(res + off);
  float4 ev = *(const float4*)(cb + (size_t)id * DDIM + lane * 4);
  float4 df = make_float4(rv.x - ev.x, rv.y - ev.y, rv.z - ev.z, rv.w - ev.w);
  float s1 = df.x * df.x + df.y * df.y + df.z * df.z + df.w * df.w;
  float s2 = ev.x * ev.x + ev.y * ev.y + ev.z * ev.z + ev.w * ev.w;
#pragma unroll
  for (int m = 16; m; m >>= 1) { s1 += __shfl_xor(s1, m, 32); s2 += __shfl_xor(s2, m, 32); }
  *(float4*)(res + off) = df;                 // res = res - emb
  float4 zv = *(float4*)(z + off);
  zv.x += ev.x; zv.y += ev.y; zv.z += ev.z; zv.w += ev.w;
  *(float4*)(z + off) = zv;                   // z += emb
  if (lane == 0) {
    qloss[b] += (1.f + CWF) * s1;             // emb_loss + CW*query_loss (equal fwd values)
    out[3 + b * NLAYERS + l] = sqrtf(s2);     // embs_norm [B,L]
  }
}

// per-row L2 normalize of sigmoid output + per-row recon partial
__global__ void norm_recon(const float* __restrict__ xh, const float* __restrict__ x,
                           float* __restrict__ reconb) {
  const int wave = threadIdx.x >> 5, lane = threadIdx.x & 31;
  const int b = blockIdx.x * 8 + wave;
  const float* row = xh + (size_t)b * INDIM;
  const float* xr = x + (size_t)b * INDIM;
  float s = 0.f;
  for (int i = lane; i < INDIM; i += 32) { const float v = row[i]; s += v * v; }
#pragma unroll
  for (int m = 16; m; m >>= 1) s += __shfl_xor(s, m, 32);
  const float inv = 1.f / fmaxf(sqrtf(s), 1e-12f);
  float acc = 0.f;
  for (int i = lane; i < INDIM; i += 32) {
    const float d = row[i] * inv - xr[i];
    acc += d * d;
  }
#pragma unroll
  for (int m = 16; m; m >>= 1) acc += __shfl_xor(acc, m, 32);
  if (lane == 0) reconb[b] = acc;
}

// deterministic single-block sum
__global__ void sum_single(const float* __restrict__ v, int n, float* __restrict__ dst) {
  __shared__ float sm[256];
  const int tid = threadIdx.x;
  float s = 0.f;
  for (int i = tid; i < n; i += 256) s += v[i];
  sm[tid] = s;
  __syncthreads();
  for (int st = 128; st; st >>= 1) {
    if (tid < st) sm[tid] += sm[tid + st];
    __syncthreads();
  }
  if (tid == 0) dst[0] = sm[0];
}

// count rows with no later duplicate (matches triu(eq,k=1) logic)
__global__ void p_unique_k(const int* __restrict__ sids, int* __restrict__ ucount) {
  const int i = blockIdx.x * blockDim.x + threadIdx.x;
  const int a0 = sids[i * 3], a1 = sids[i * 3 + 1], a2 = sids[i * 3 + 2];
  int dup = 0;
  for (int j = i + 1; j < BDIM; ++j) {
    if (sids[j * 3] == a0 && sids[j * 3 + 1] == a1 && sids[j * 3 + 2] == a2) { dup = 1; break; }
  }
  if (!dup) atomicAdd(ucount, 1);
}

__global__ void finalize_k(const float* __restrict__ recon_tot, const float* __restrict__ qsum,
                           const int* __restrict__ ucount, float* __restrict__ out) {
  const float recon = recon_tot[0];
  const float qmean = qsum[0] / (float)BDIM;
  out[0] = recon + qmean;   // loss = (recon + qloss).mean()
  out[1] = recon;
  out[2] = qmean;
  out[3 + BDIM * NLAYERS] = (float)ucount[0] / (float)BDIM;
}

extern "C" void kernel_launch(void* const* d_in, const int* in_sizes, int n_in,
                              void* d_out, int out_size, void* d_ws, size_t ws_size,
                              hipStream_t stream) {
  (void)in_sizes; (void)n_in; (void)out_size; (void)ws_size;
  const float* x   = (const float*)d_in[0];
  const float* ew1 = (const float*)d_in[1];  const float* eb1 = (const float*)d_in[2];
  const float* ew2 = (const float*)d_in[3];  const float* eb2 = (const float*)d_in[4];
  const float* ew3 = (const float*)d_in[5];  const float* eb3 = (const float*)d_in[6];
  const float* dw1 = (const float*)d_in[7];  const float* db1 = (const float*)d_in[8];
  const float* dw2 = (const float*)d_in[9];  const float* db2 = (const float*)d_in[10];
  const float* dw3 = (const float*)d_in[11]; const float* db3 = (const float*)d_in[12];
  const float* codebooks = (const float*)d_in[13];
  const float* projs     = (const float*)d_in[14];
  float* out = (float*)d_out;

  // workspace layout (~64 MB)
  float* ws  = (float*)d_ws;
  float* h1  = ws;                              // [B,512]  (enc h1, later dec g2)
  float* h2  = h1 + (size_t)BDIM * 512;         // [B,256]  (enc h2, later dec g1)
  float* res = h2 + (size_t)BDIM * 256;         // [B,128]
  float* z   = res + (size_t)BDIM * DDIM;       // [B,128]
  float* cb  = z + (size_t)BDIM * DDIM;         // [K,128]  projected codebook
  float* xh  = cb + (size_t)KCB * DDIM;         // [B,768]
  float* cbsq   = xh + (size_t)BDIM * INDIM;    // [K]
  float* qlossb = cbsq + KCB;                   // [B]
  float* reconb = qlossb + BDIM;                // [B]
  float* pscore = reconb + BDIM;                // [NSEG,B]
  int*   pidx   = (int*)(pscore + (size_t)NSEG * BDIM);  // [NSEG,B]
  int*   sids   = pidx + (size_t)NSEG * BDIM;            // [B,3]
  float* recon_tot = (float*)(sids + (size_t)BDIM * NLAYERS);
  float* qsum   = recon_tot + 1;
  int*   ucount = (int*)(qsum + 1);

  hipMemsetAsync(z, 0, (size_t)BDIM * DDIM * sizeof(float), stream);
  hipMemsetAsync(qlossb, 0, (size_t)BDIM * sizeof(float), stream);
  hipMemsetAsync(recon_tot, 0, 3 * sizeof(float), stream);

  const dim3 blk(256);

  // encoder
  mlp_gemm<<<dim3(BDIM / 128, 512 / 16), blk, 0, stream>>>(x, ew1, eb1, h1, 768, 512, 1);
  mlp_gemm<<<dim3(BDIM / 128, 256 / 16), blk, 0, stream>>>(h1, ew2, eb2, h2, 512, 256, 1);
  mlp_gemm<<<dim3(BDIM / 128, 128 / 16), blk, 0, stream>>>(h2, ew3, eb3, res, 256, 128, 0);

  // residual VQ layers
  for (int l = 0; l < NLAYERS; ++l) {
    mlp_gemm<<<dim3(KCB / 128, 128 / 16), blk, 0, stream>>>(
        codebooks + (size_t)l * KCB * DDIM, projs + (size_t)l * DDIM * DDIM,
        nullptr, cb, 128, 128, 0);
    row_sq<<<dim3(KCB / 8), blk, 0, stream>>>(cb, cbsq);
    dist_argmin<<<dim3(BDIM / 128, NSEG), blk, 0, stream>>>(res, cb, cbsq, pscore, pidx);
    argmin_fin<<<dim3(BDIM / 256), blk, 0, stream>>>(pscore, pidx, sids, l);
    vq_update<<<dim3(BDIM / 8), blk, 0, stream>>>(res, cb, sids, z, qlossb, out, l);
  }

  // decoder (reuse h2 as g1, h1 as g2)
  mlp_gemm<<<dim3(BDIM / 128, 256 / 16), blk, 0, stream>>>(z, dw1, db1, h2, 128, 256, 1);
  mlp_gemm<<<dim3(BDIM / 128, 512 / 16), blk, 0, stream>>>(h2, dw2, db2, h1, 256, 512, 1);
  mlp_gemm<<<dim3(BDIM / 128, 768 / 16), blk, 0, stream>>>(h1, dw3, db3, xh, 512, 768, 2);

  norm_recon<<<dim3(BDIM / 8), blk, 0, stream>>>(xh, x, reconb);
  sum_single<<<dim3(1), blk, 0, stream>>>(reconb, BDIM, recon_tot);
  sum_single<<<dim3(1), blk, 0, stream>>>(qlossb, BDIM, qsum);
  p_unique_k<<<dim3(BDIM / 256), blk, 0, stream>>>(sids, ucount);
  finalize_k<<<dim3(1), dim3(1), 0, stream>>>(recon_tot, qsum, ucount, out);
}